// E3Hamiltonian_70866960384254
// MI455X (gfx1250) — compile-verified
//
#include <hip/hip_runtime.h>

typedef float v2f __attribute__((ext_vector_type(2)));
typedef float v8f __attribute__((ext_vector_type(8)));

#define NCOL 484
#define WAVES_PER_BLOCK 2
#define TILE_BYTES (16 * NCOL * 4)  // 30976 B per 16-row tile
#define REGION_FLOATS 7748          // padded: covers over-reads, 16B-multiple

// Pairtype slice offsets within a 484-wide row
#define OFF_SS 0    // c=9,  m=1
#define OFF_SP 9    // c=18, m=3
#define OFF_SD 63   // c=12, m=5
#define OFF_PP 123  // c=9,  m=9
#define OFF_PD 204  // c=12, m=15
#define OFF_DD 384  // c=4,  m=25

#define Z8 {0.f, 0.f, 0.f, 0.f, 0.f, 0.f, 0.f, 0.f}

// ---------------- gfx1250 async global<->LDS copy (ASYNCcnt path) -----------
// INST_OFFSET is added to BOTH the LDS and global addresses (ISA 15.18), so a
// linear tile copy is one addr-VGPR set + immediate offsets.
template <int IT, int END>
struct CopyIn {
  static __device__ __forceinline__ void run(unsigned l, const char* g) {
    asm volatile("global_load_async_to_lds_b128 %0, %1, off offset:%2"
                 :: "v"(l), "v"(g), "i"(IT * 512) : "memory");
    CopyIn<IT + 1, END>::run(l, g);
  }
};
template <int END>
struct CopyIn<END, END> {
  static __device__ __forceinline__ void run(unsigned, const char*) {}
};

template <int IT, int END>
struct CopyOut {
  static __device__ __forceinline__ void run(char* g, unsigned l) {
    asm volatile("global_store_async_from_lds_b128 %0, %1, off offset:%2"
                 :: "v"(g), "v"(l), "i"(IT * 512) : "memory");
    CopyOut<IT + 1, END>::run(g, l);
  }
};
template <int END>
struct CopyOut<END, END> {
  static __device__ __forceinline__ void run(char*, unsigned) {}
};

static __device__ __forceinline__ void async_load_b64(unsigned lds, const void* g) {
  asm volatile("global_load_async_to_lds_b64 %0, %1, off" :: "v"(lds), "v"(g) : "memory");
}
static __device__ __forceinline__ void async_store_b64(void* g, unsigned lds) {
  asm volatile("global_store_async_from_lds_b64 %0, %1, off" :: "v"(g), "v"(lds) : "memory");
}
static __device__ __forceinline__ void wait_asynccnt0() {
  asm volatile("s_wait_asynccnt 0x0" ::: "memory");
}
static __device__ __forceinline__ void wait_dscnt0() {
  asm volatile("s_wait_dscnt 0x0" ::: "memory");
}

// ---------------- WMMA helpers ----------------------------------------------
static __device__ __forceinline__ v8f wmma4(v2f a, v2f b, v8f c) {
  return __builtin_amdgcn_wmma_f32_16x16x4_f32(
      false, a, false, b, (short)0, c, false, false);
}

// Branchless B fragment: B[k,n] = cg[n*m+k], zero-padded via cndmask
static __device__ __forceinline__ v2f bfrag(const float* __restrict__ cg, int m,
                                            int kk, int t, int half, int lrow) {
  int n  = t * 16 + lrow;
  int k0 = kk * 4 + 2 * half;
  int nc  = (n < m) ? n : (m - 1);
  int kc0 = (k0 < m) ? k0 : (m - 1);
  int kc1 = (k0 + 1 < m) ? (k0 + 1) : (m - 1);
  float v0 = cg[nc * m + kc0];
  float v1 = cg[nc * m + kc1];
  v2f b;
  b.x = (n < m && k0 < m)     ? v0 : 0.f;
  b.y = (n < m && k0 + 1 < m) ? v1 : 0.f;
  return b;
}

// Branchless A gather from this lane's LDS row (region padded -> over-reads ok)
static __device__ __forceinline__ v2f afrag(const float* rowp, int m, int base,
                                            int kk, int half) {
  int k0 = kk * 4 + 2 * half;
  float a0 = rowp[base + k0];
  float a1 = rowp[base + k0 + 1];
  v2f a;
  a.x = (k0 < m)     ? a0 : 0.f;
  a.y = (k0 + 1 < m) ? a1 : 0.f;
  return a;
}

// D scatter into the LDS tile, in place over the chunk's input slot
static __device__ __forceinline__ void store_d(v8f d, float* tilebase, int col,
                                               int nvalid, int lrow, int half) {
  if (lrow < nvalid) {
#pragma unroll
    for (int v = 0; v < 8; ++v)
      tilebase[(v + 8 * half) * NCOL + col + lrow] = d[v];
  }
}

// pd is register-heavy: process 6 chunks per batch (loads first, then compute)
static __device__ __forceinline__ void pd_batch(float* my, const float* rowp,
                                                const v2f* b_pd, int ci0,
                                                int lrow, int half) {
  v2f a[6][4];
#pragma unroll
  for (int c = 0; c < 6; ++c)
#pragma unroll
    for (int kk = 0; kk < 4; ++kk)
      a[c][kk] = afrag(rowp, 15, OFF_PD + (ci0 + c) * 15, kk, half);
#pragma unroll
  for (int c = 0; c < 6; ++c) {
    v8f acc = Z8;
#pragma unroll
    for (int kk = 0; kk < 4; ++kk) acc = wmma4(a[c][kk], b_pd[kk], acc);
    store_d(acc, my, OFF_PD + (ci0 + c) * 15, 15, lrow, half);
  }
}

// ---------------- main kernel: one full 16-row tile per wave ----------------
__global__ __launch_bounds__(32 * WAVES_PER_BLOCK) void e3_cg_wmma_kernel(
    const float* __restrict__ x, float* __restrict__ y, int ntiles,
    const float* __restrict__ cg_ss, const float* __restrict__ cg_sp,
    const float* __restrict__ cg_sd, const float* __restrict__ cg_pp,
    const float* __restrict__ cg_pd, const float* __restrict__ cg_dd) {
  __shared__ float smem[WAVES_PER_BLOCK * REGION_FLOATS];

  const int lane = threadIdx.x & 31;
  const int wave = threadIdx.x >> 5;
  const int half = lane >> 4;
  const int lrow = lane & 15;

  const int tile = blockIdx.x * WAVES_PER_BLOCK + wave;
  if (tile >= ntiles) return;  // wave-uniform: EXEC all-ones for WMMA
  const int row0 = tile * 16;

  float* my = smem + wave * REGION_FLOATS;
  const unsigned lds_base = (unsigned)(size_t)my;  // low 32 bits = LDS offset

  // ---- async copy-in: contiguous tile, 60 x b128 (+1 x b64), shared addrs
  const char* gsrc = (const char*)(x + (size_t)row0 * NCOL);
  CopyIn<0, 60>::run(lds_base + (unsigned)lane * 16u, gsrc + lane * 16);
  async_load_b64(lds_base + 30720u + (unsigned)lane * 8u, gsrc + 30720 + lane * 8);

  // ---- B fragments while the copy is in flight (tiny, L2-resident) ----
  v2f b_sp0 = bfrag(cg_sp, 3, 0, 0, half, lrow);
  v2f b_sd[2], b_pp[3], b_pd[4], b_dd0[7], b_dd1[7];
#pragma unroll
  for (int kk = 0; kk < 2; ++kk) b_sd[kk] = bfrag(cg_sd, 5, kk, 0, half, lrow);
#pragma unroll
  for (int kk = 0; kk < 3; ++kk) b_pp[kk] = bfrag(cg_pp, 9, kk, 0, half, lrow);
#pragma unroll
  for (int kk = 0; kk < 4; ++kk) b_pd[kk] = bfrag(cg_pd, 15, kk, 0, half, lrow);
#pragma unroll
  for (int kk = 0; kk < 7; ++kk) {
    b_dd0[kk] = bfrag(cg_dd, 25, kk, 0, half, lrow);
    b_dd1[kk] = bfrag(cg_dd, 25, kk, 1, half, lrow);
  }
  const float s_ss = cg_ss[0];

  wait_asynccnt0();  // tile now in LDS

  float* rowp = my + lrow * NCOL;  // this lane's A row (lanes 16-31 mirror rows)

  // ---- ss: m=1 scale in place; halves cover cols 0-4 / 4-8 (col 4 written
  // twice with the identical value; all loads precede all stores) ----
  {
    const int c0 = half * 4;
    float t[5];
#pragma unroll
    for (int i = 0; i < 5; ++i) t[i] = rowp[c0 + i];
#pragma unroll
    for (int i = 0; i < 5; ++i) rowp[c0 + i] = s_ss * t[i];
  }

  // ---- sp: m=3, c=18, 1 K-step (all loads, then compute+store) ----
  {
    v2f a[18];
#pragma unroll
    for (int ci = 0; ci < 18; ++ci) a[ci] = afrag(rowp, 3, OFF_SP + ci * 3, 0, half);
#pragma unroll
    for (int ci = 0; ci < 18; ++ci) {
      v8f acc = Z8;
      acc = wmma4(a[ci], b_sp0, acc);
      store_d(acc, my, OFF_SP + ci * 3, 3, lrow, half);
    }
  }

  // ---- sd: m=5, c=12, 2 K-steps ----
  {
    v2f a[12][2];
#pragma unroll
    for (int ci = 0; ci < 12; ++ci)
#pragma unroll
      for (int kk = 0; kk < 2; ++kk)
        a[ci][kk] = afrag(rowp, 5, OFF_SD + ci * 5, kk, half);
#pragma unroll
    for (int ci = 0; ci < 12; ++ci) {
      v8f acc = Z8;
#pragma unroll
      for (int kk = 0; kk < 2; ++kk) acc = wmma4(a[ci][kk], b_sd[kk], acc);
      store_d(acc, my, OFF_SD + ci * 5, 5, lrow, half);
    }
  }

  // ---- pp: m=9, c=9, 3 K-steps ----
  {
    v2f a[9][3];
#pragma unroll
    for (int ci = 0; ci < 9; ++ci)
#pragma unroll
      for (int kk = 0; kk < 3; ++kk)
        a[ci][kk] = afrag(rowp, 9, OFF_PP + ci * 9, kk, half);
#pragma unroll
    for (int ci = 0; ci < 9; ++ci) {
      v8f acc = Z8;
#pragma unroll
      for (int kk = 0; kk < 3; ++kk) acc = wmma4(a[ci][kk], b_pp[kk], acc);
      store_d(acc, my, OFF_PP + ci * 9, 9, lrow, half);
    }
  }

  // ---- pd: m=15, c=12, 4 K-steps; two 6-chunk batches (VGPR bound) ----
  pd_batch(my, rowp, b_pd, 0, lrow, half);
  pd_batch(my, rowp, b_pd, 6, lrow, half);

  // ---- dd: m=25, c=4, 7 K-steps, 2 N-tiles sharing A ----
  {
    v2f a[4][7];
#pragma unroll
    for (int ci = 0; ci < 4; ++ci)
#pragma unroll
      for (int kk = 0; kk < 7; ++kk)
        a[ci][kk] = afrag(rowp, 25, OFF_DD + ci * 25, kk, half);
#pragma unroll
    for (int ci = 0; ci < 4; ++ci) {
      v8f acc0 = Z8, acc1 = Z8;
#pragma unroll
      for (int kk = 0; kk < 7; ++kk) {
        acc0 = wmma4(a[ci][kk], b_dd0[kk], acc0);
        acc1 = wmma4(a[ci][kk], b_dd1[kk], acc1);
      }
      store_d(acc0, my, OFF_DD + ci * 25, 16, lrow, half);
      store_d(acc1, my, OFF_DD + ci * 25 + 16, 9, lrow, half);
    }
  }

  // ---- async copy-out: LDS tile -> contiguous global, coalesced b128 ----
  wait_dscnt0();  // all ds_stores visible before the async engine reads LDS
  char* gdst = (char*)(y + (size_t)row0 * NCOL);
  CopyOut<0, 60>::run(gdst + lane * 16, lds_base + (unsigned)lane * 16u);
  async_store_b64(gdst + 30720 + lane * 8, lds_base + 30720u + (unsigned)lane * 8u);
  // s_endpgm performs an implicit wait-idle (ASYNCcnt drains before exit)
}

// ---------------- scalar tail kernel: <=15 leftover rows --------------------
__global__ void e3_cg_tail_kernel(const float* __restrict__ x, float* __restrict__ y,
                                  int row_start,
                                  const float* __restrict__ cg_ss, const float* __restrict__ cg_sp,
                                  const float* __restrict__ cg_sd, const float* __restrict__ cg_pp,
                                  const float* __restrict__ cg_pd, const float* __restrict__ cg_dd) {
  const int row = row_start + blockIdx.x;
  const float* xr = x + (size_t)row * NCOL;
  float* yr = y + (size_t)row * NCOL;
  for (int col = threadIdx.x; col < NCOL; col += blockDim.x) {
    float acc = 0.f;
    if (col < OFF_SP) {
      acc = cg_ss[0] * xr[col];
    } else if (col < OFF_SD) {
      int o = col - OFF_SP, li = o % 3, b = OFF_SP + (o / 3) * 3;
      for (int k = 0; k < 3; ++k) acc += cg_sp[li * 3 + k] * xr[b + k];
    } else if (col < OFF_PP) {
      int o = col - OFF_SD, li = o % 5, b = OFF_SD + (o / 5) * 5;
      for (int k = 0; k < 5; ++k) acc += cg_sd[li * 5 + k] * xr[b + k];
    } else if (col < OFF_PD) {
      int o = col - OFF_PP, li = o % 9, b = OFF_PP + (o / 9) * 9;
      for (int k = 0; k < 9; ++k) acc += cg_pp[li * 9 + k] * xr[b + k];
    } else if (col < OFF_DD) {
      int o = col - OFF_PD, li = o % 15, b = OFF_PD + (o / 15) * 15;
      for (int k = 0; k < 15; ++k) acc += cg_pd[li * 15 + k] * xr[b + k];
    } else {
      int o = col - OFF_DD, li = o % 25, b = OFF_DD + (o / 25) * 25;
      for (int k = 0; k < 25; ++k) acc += cg_dd[li * 25 + k] * xr[b + k];
    }
    yr[col] = acc;
  }
}

extern "C" void kernel_launch(void* const* d_in, const int* in_sizes, int n_in,
                              void* d_out, int out_size, void* d_ws, size_t ws_size,
                              hipStream_t stream) {
  const float* edge_x = (const float*)d_in[0];
  const float* node_x = (const float*)d_in[1];
  // d_in[2] = edge_index (int64) — unused by the reference computation
  const float* cg_ss = (const float*)d_in[3];
  const float* cg_sp = (const float*)d_in[4];
  const float* cg_sd = (const float*)d_in[5];
  const float* cg_pp = (const float*)d_in[6];
  const float* cg_pd = (const float*)d_in[7];
  const float* cg_dd = (const float*)d_in[8];

  const int n_edge = in_sizes[0] / NCOL;
  const int n_node = in_sizes[1] / NCOL;

  float* y_edge = (float*)d_out;
  float* y_node = (float*)d_out + (size_t)n_edge * NCOL;

  const int te = n_edge >> 4, re = n_edge & 15;   // 400000 -> 25000 tiles, 0 tail
  const int tn = n_node >> 4, rn = n_node & 15;   // 25000  -> 1562 tiles, 8 tail

  const int thr = 32 * WAVES_PER_BLOCK;
  if (te)
    e3_cg_wmma_kernel<<<(te + WAVES_PER_BLOCK - 1) / WAVES_PER_BLOCK, thr, 0, stream>>>(
        edge_x, y_edge, te, cg_ss, cg_sp, cg_sd, cg_pp, cg_pd, cg_dd);
  if (tn)
    e3_cg_wmma_kernel<<<(tn + WAVES_PER_BLOCK - 1) / WAVES_PER_BLOCK, thr, 0, stream>>>(
        node_x, y_node, tn, cg_ss, cg_sp, cg_sd, cg_pp, cg_pd, cg_dd);
  if (re)
    e3_cg_tail_kernel<<<re, 128, 0, stream>>>(edge_x, y_edge, te * 16,
                                              cg_ss, cg_sp, cg_sd, cg_pp, cg_pd, cg_dd);
  if (rn)
    e3_cg_tail_kernel<<<rn, 128, 0, stream>>>(node_x, y_node, tn * 16,
                                              cg_ss, cg_sp, cg_sd, cg_pp, cg_pd, cg_dd);
}